// SelfAttention_50431505990250
// MI455X (gfx1250) — compile-verified
//
#include <hip/hip_runtime.h>

// Problem constants (match reference)
constexpr int kB = 8;
constexpr int kL = 4096;
constexpr int kD = 768;
constexpr float kEps = 1e-7f;
// ATTN_WIDTH = 12 -> band k in [m+5, m+16] inside a 32-wide j-slice [r-16, r+15]

typedef __attribute__((ext_vector_type(2))) float v2f;
typedef __attribute__((ext_vector_type(8))) float v8f;

// ---------------------------------------------------------------------------
// Kernel 1: u[b,l] = x[b,l,:] . Wa   (wave per row, float4 coalesced loads)
// ---------------------------------------------------------------------------
__global__ __launch_bounds__(256) void compute_u_kernel(
    const float* __restrict__ x, const float* __restrict__ Wa,
    float* __restrict__ u) {
  const int wave = threadIdx.x >> 5;                 // 0..7
  const int lane = threadIdx.x & 31;
  const int row  = blockIdx.x * 8 + wave;            // flattened b*L + l
  const float* xr = x + (size_t)row * kD;

  float acc = 0.0f;
#pragma unroll
  for (int it = 0; it < kD / (32 * 4); ++it) {       // 6 iterations
    const int d = (it * 32 + lane) * 4;
    const float4 xv = *(const float4*)(xr + d);
    const float4 wv = *(const float4*)(Wa + d);
    acc += xv.x * wv.x + xv.y * wv.y + xv.z * wv.z + xv.w * wv.w;
  }
  // wave32 butterfly reduction
#pragma unroll
  for (int off = 16; off > 0; off >>= 1)
    acc += __shfl_xor(acc, off, 32);
  if (lane == 0) u[row] = acc;
}

// ---------------------------------------------------------------------------
// Kernel 2: banded matmul via V_WMMA_F32_16X16X4_F32.
// One workgroup (4 waves) per 16-row tile; each wave covers 12 of the 48
// 16-column tiles of D=768.  A(16x32) is the banded u-matrix, B is the x
// slice for j in [r-16, r+15].
// ---------------------------------------------------------------------------
__global__ __launch_bounds__(128) void banded_attn_kernel(
    const float* __restrict__ x, const float* __restrict__ u,
    float* __restrict__ out) {
  const int nblk = kL / 16;                          // 256 row tiles per batch
  const int b    = blockIdx.x / nblk;
  const int r    = (blockIdx.x % nblk) * 16;         // within-batch row of tile
  const int tid  = threadIdx.x;
  const int lane = tid & 31;
  const int wave = tid >> 5;

  __shared__ float u_loc[32];     // u[r-16 .. r+15], 0 for j<0
  __shared__ float scale_s[16];   // u_i / (u_i*wsum_i + eps) per tile row

  if (tid < 32) {
    const int j = r - 16 + tid;
    u_loc[tid] = (j >= 0) ? u[(size_t)b * kL + j] : 0.0f;
  }
  __syncthreads();
  if (tid < 16) {
    const float ui = u_loc[16 + tid];
    float ws = 0.0f;
#pragma unroll
    for (int k = tid + 5; k <= tid + 16; ++k) ws += u_loc[k];  // 12-wide window
    scale_s[tid] = ui / (ui * ws + kEps);
  }
  __syncthreads();

  // A fragment layout (32-bit A 16x4): lanes 0-15 hold M=0-15 / K={0,1},
  // lanes 16-31 hold M=0-15 / K={2,3}; VGPR0 <-> K even, VGPR1 <-> K odd.
  const int m     = lane & 15;
  const int khalf = (lane >> 4) * 2;                 // 0 or 2
  float a_reg[8][2];
#pragma unroll
  for (int kc = 0; kc < 8; ++kc) {
#pragma unroll
    for (int vv = 0; vv < 2; ++vv) {
      const int k = 4 * kc + khalf + vv;             // 0..31
      a_reg[kc][vv] = (k >= m + 5 && k <= m + 16) ? u_loc[k] : 0.0f;
    }
  }

  const size_t xbase = (size_t)b * kL * kD;

  for (int t = 0; t < 12; ++t) {                     // uniform across waves
    const int d0 = (wave * 12 + t) * 16;
    v8f c = {};
#pragma unroll
    for (int kc = 0; kc < 8; ++kc) {
      v2f av;
      av[0] = a_reg[kc][0];
      av[1] = a_reg[kc][1];
      v2f bv;                                        // B 4x16, K layout mirrors A
#pragma unroll
      for (int vv = 0; vv < 2; ++vv) {
        const int k  = 4 * kc + khalf + vv;
        const int j  = r - 16 + k;
        const int jc = (j < 0) ? 0 : j;              // A==0 there, value unused
        bv[vv] = x[xbase + (size_t)jc * kD + d0 + m];
      }
      c = __builtin_amdgcn_wmma_f32_16x16x4_f32(false, av, false, bv,
                                                (short)0, c, false, false);
    }
    // C/D 16x16 f32 layout: VGPR t -> M=t (lanes 0-15) / M=t+8 (lanes 16-31)
    const int mbase = (lane >> 4) * 8;
#pragma unroll
    for (int tt = 0; tt < 8; ++tt) {
      const int mm = mbase + tt;
      out[xbase + (size_t)(r + mm) * kD + d0 + m] = c[tt] * scale_s[mm];
    }
  }
}

// ---------------------------------------------------------------------------
extern "C" void kernel_launch(void* const* d_in, const int* in_sizes, int n_in,
                              void* d_out, int out_size, void* d_ws,
                              size_t ws_size, hipStream_t stream) {
  const float* x  = (const float*)d_in[0];   // [B, L, D] f32
  const float* Wa = (const float*)d_in[1];   // [D, 1]    f32
  float* out = (float*)d_out;                // [B, L, D] f32
  float* u   = (float*)d_ws;                 // [B*L] f32 scratch (128 KB)

  compute_u_kernel<<<(kB * kL) / 8, 256, 0, stream>>>(x, Wa, u);
  banded_attn_kernel<<<kB * (kL / 16), 128, 0, stream>>>(x, u, out);
}